// ProbabilisticSurfaceDistanceLoss_60292750901759
// MI455X (gfx1250) — compile-verified
//
#include <hip/hip_runtime.h>

// ---------------------------------------------------------------------------
// ProbabilisticSurfaceDistanceLoss on MI455X (gfx1250)
// d2(a,b) = |a|^2 + dot4((ax,ay,az,1), (-2bx,-2by,-2bz,|b|^2))
// -> rank-4 GEMM tiles on V_WMMA_F32_16X16X4_F32 (f32, matches reference).
// B tiles streamed through double-buffered LDS using async global->LDS DMA
// (ASYNCcnt) so WMMA overlaps the staging copy; B range split 8 ways for wave
// parallelism (B tile count padded to NSPLIT*4 so every split runs only the
// 4x-unrolled WMMA loop, no scalar tail); associative min combined in a
// second pass. Fully deterministic (no float atomics).
// ---------------------------------------------------------------------------

typedef __attribute__((ext_vector_type(2))) float v2f;   // A/B operand: 16x4 f32 = 2 VGPRs
typedef __attribute__((ext_vector_type(8))) float v8f;   // C/D operand: 16x16 f32 = 8 VGPRs
typedef __attribute__((ext_vector_type(4))) int   v4i;   // b128 async-copy payload

#define EPS_F        1e-8f
#define PROB_PEN_F   1e-4f
#define REV_SCALE_F  0.1f
#define BIG_F        3.0e38f
#define CHUNK        64      // B tiles per LDS buffer (64*256B = 16KB; x2 buffers)
#define NSPLIT       8       // B-range splits (gridDim.y)
#define TILE_ALIGN   (NSPLIT * 4)   // pad B tile count: per-split count % 4 == 0

#if __has_builtin(__builtin_amdgcn_global_load_async_to_lds_b128)
#define ASYNC_LDS 1
#else
#define ASYNC_LDS 0
#endif

__device__ __forceinline__ float waveReduceSum(float v) {
#pragma unroll
  for (int off = 16; off > 0; off >>= 1) v += __shfl_down(v, off, 32);
  return v;
}

__device__ __forceinline__ float waveReduceMax(float v) {
#pragma unroll
  for (int off = 16; off > 0; off >>= 1) v = fmaxf(v, __shfl_down(v, off, 32));
  return v;
}

// Stage nFloat4 float4's from global to LDS (async DMA, ASYNCcnt-tracked).
__device__ __forceinline__ void stage_issue(const float* __restrict__ gsrc,
                                            float* sdst, int nFloat4, int tid) {
  const float4* g4 = (const float4*)gsrc;
#if ASYNC_LDS
  for (int idx = tid; idx < nFloat4; idx += 256) {
    __builtin_amdgcn_global_load_async_to_lds_b128(
        (v4i*)(g4 + idx), (v4i*)((float4*)sdst + idx), 0, 0);
  }
#else
  float4* s4 = (float4*)sdst;
  for (int idx = tid; idx < nFloat4; idx += 256) s4[idx] = g4[idx];
#endif
}

__device__ __forceinline__ void stage_wait() {
#if ASYNC_LDS
#if __has_builtin(__builtin_amdgcn_s_wait_asynccnt)
  __builtin_amdgcn_s_wait_asynccnt(0);
#else
  asm volatile("s_wait_asynccnt 0x0" ::: "memory");
#endif
#endif
}

// ---------------------------------------------------------------------------
__global__ void k_init(unsigned* max_bits) {
  if (threadIdx.x == 0) max_bits[0] = 0u;
}

// ---------------------------------------------------------------------------
// B-tile packing: per tile t (64 floats), lane l holds v2f at [t*64 + l*2],
// value(K = (l<16 ? j : 2+j), N = l%16), j=0,1. Column n = (-2x,-2y,-2z,|b|^2);
// padded columns (incl. whole sentinel tiles) get |b|^2 = +BIG.
// ---------------------------------------------------------------------------
__global__ void k_pack_orig_bc(const float* __restrict__ V, const int* __restrict__ F,
                               int Fo, int FoPad, float* __restrict__ Bbuf) {
  int i = blockIdx.x * blockDim.x + threadIdx.x;
  if (i >= FoPad) return;
  int tile = i >> 4, n = i & 15;
  float x = 0.f, y = 0.f, z = 0.f, nn = BIG_F;
  if (i < Fo) {
    int a = F[3 * i], b = F[3 * i + 1], c = F[3 * i + 2];
    x = (V[3 * a] + V[3 * b] + V[3 * c]) * (1.f / 3.f);
    y = (V[3 * a + 1] + V[3 * b + 1] + V[3 * c + 1]) * (1.f / 3.f);
    z = (V[3 * a + 2] + V[3 * b + 2] + V[3 * c + 2]) * (1.f / 3.f);
    nn = x * x + y * y + z * z;
  }
  float* base = Bbuf + (size_t)tile * 64;
  base[n * 2 + 0]        = -2.f * x;   // K=0
  base[n * 2 + 1]        = -2.f * y;   // K=1
  base[(16 + n) * 2 + 0] = -2.f * z;   // K=2
  base[(16 + n) * 2 + 1] = nn;         // K=3
}

__global__ void k_pack_verts(const float* __restrict__ V, int No, int NoPad,
                             float* __restrict__ Bbuf) {
  int i = blockIdx.x * blockDim.x + threadIdx.x;
  if (i >= NoPad) return;
  int tile = i >> 4, n = i & 15;
  float x = 0.f, y = 0.f, z = 0.f, nn = BIG_F;
  if (i < No) {
    x = V[3 * i]; y = V[3 * i + 1]; z = V[3 * i + 2];
    nn = x * x + y * y + z * z;
  }
  float* base = Bbuf + (size_t)tile * 64;
  base[n * 2 + 0]        = -2.f * x;
  base[n * 2 + 1]        = -2.f * y;
  base[(16 + n) * 2 + 0] = -2.f * z;
  base[(16 + n) * 2 + 1] = nn;
}

// ---------------------------------------------------------------------------
// A-tile packing: lane l holds v2f at [t*64 + l*2], value(M = l%16,
// K = (l<16 ? j : 2+j)); K=3 component = 1.0 (picks up |b|^2 from B).
// Row norms go to nA. Also wave-reduces sum(fp) into partials.
// ---------------------------------------------------------------------------
__global__ void k_pack_simp_bc(const float* __restrict__ V, const int* __restrict__ F,
                               const float* __restrict__ fp, int Fs, int FsPad,
                               float* __restrict__ Abuf, float* __restrict__ nA,
                               float* __restrict__ fpPart) {
  int i = blockIdx.x * blockDim.x + threadIdx.x;
  float p = 0.f;
  if (i < FsPad) {
    int tile = i >> 4, m = i & 15;
    float x = 0.f, y = 0.f, z = 0.f;
    if (i < Fs) {
      int a = F[3 * i], b = F[3 * i + 1], c = F[3 * i + 2];
      x = (V[3 * a] + V[3 * b] + V[3 * c]) * (1.f / 3.f);
      y = (V[3 * a + 1] + V[3 * b + 1] + V[3 * c + 1]) * (1.f / 3.f);
      z = (V[3 * a + 2] + V[3 * b + 2] + V[3 * c + 2]) * (1.f / 3.f);
      p = fp[i];
    }
    float* base = Abuf + (size_t)tile * 64;
    base[m * 2 + 0]        = x;
    base[m * 2 + 1]        = y;
    base[(16 + m) * 2 + 0] = z;
    base[(16 + m) * 2 + 1] = 1.0f;
    nA[i] = x * x + y * y + z * z;
  }
  float s = waveReduceSum(p);
  int w = (blockIdx.x * blockDim.x + threadIdx.x) >> 5;
  if ((threadIdx.x & 31) == 0) fpPart[w] = s;
}

__global__ void k_pack_samples(const float* __restrict__ SV, const int* __restrict__ SF,
                               const float* __restrict__ u1, const float* __restrict__ u2,
                               int Fs, int S, int P, int PPad,
                               float* __restrict__ Abuf, float* __restrict__ nS) {
  int i = blockIdx.x * blockDim.x + threadIdx.x;
  if (i >= PPad) return;
  int tile = i >> 4, m = i & 15;
  float x = 0.f, y = 0.f, z = 0.f;
  if (i < P) {
    int f = i / S, s = i - f * S;
    float r1 = sqrtf(u1[f * S + s]);
    float uu = u2[f * S + s];
    float wa = 1.f - r1, wb = r1 * (1.f - uu), wc = r1 * uu;
    int i0 = SF[3 * f], i1 = SF[3 * f + 1], i2 = SF[3 * f + 2];
    x = wa * SV[3 * i0] + wb * SV[3 * i1] + wc * SV[3 * i2];
    y = wa * SV[3 * i0 + 1] + wb * SV[3 * i1 + 1] + wc * SV[3 * i2 + 1];
    z = wa * SV[3 * i0 + 2] + wb * SV[3 * i1 + 2] + wc * SV[3 * i2 + 2];
  }
  float* base = Abuf + (size_t)tile * 64;
  base[m * 2 + 0]        = x;
  base[m * 2 + 1]        = y;
  base[(16 + m) * 2 + 0] = z;
  base[(16 + m) * 2 + 1] = 1.0f;
  nS[i] = x * x + y * y + z * z;
}

// ---------------------------------------------------------------------------
// Double-buffered LDS-staged WMMA row-min. Per chunk: (1) wait async DMA +
// barrier (publishes chunk c AND guarantees chunk c-1's buffer is consumed),
// (2) issue async DMA for chunk c+1 into the other buffer, (3) WMMAs off LDS.
// cntTiles is a multiple of 4 -> only the 4x-unrolled loop is executed.
// D layout: VGPR v, lanes 0-15 -> M=v, N=lane; lanes 16-31 -> M=v+8, N=lane-16.
// ---------------------------------------------------------------------------
__device__ __forceinline__ void staged_row_min(const float* __restrict__ Abuf,
                                               const float* __restrict__ Bbuf,
                                               int atile, int baseTile, int cntTiles,
                                               float* smB /* [2][CHUNK*64] */,
                                               int lane, int tid, v8f& rmin) {
  v2f a = ((const v2f*)(Abuf + (size_t)atile * 64))[lane];
#pragma unroll
  for (int k = 0; k < 8; ++k) rmin[k] = BIG_F;
  int nChunks = (cntTiles + CHUNK - 1) / CHUNK;
  if (nChunks > 0) {
    int n0 = cntTiles < CHUNK ? cntTiles : CHUNK;
    stage_issue(Bbuf + (size_t)baseTile * 64, smB, n0 * 16, tid);
  }
  int buf = 0;
  for (int c = 0; c < nChunks; ++c, buf ^= 1) {
    stage_wait();        // chunk c resident in smB[buf]
    __syncthreads();     // ...visible to all waves; buf^1 fully consumed
    if (c + 1 < nChunks) {
      int nn = cntTiles - (c + 1) * CHUNK;
      if (nn > CHUNK) nn = CHUNK;
      stage_issue(Bbuf + (size_t)(baseTile + (c + 1) * CHUNK) * 64,
                  smB + (buf ^ 1) * (CHUNK * 64), nn * 16, tid);
    }
    int n = cntTiles - c * CHUNK;      // always a multiple of 4
    if (n > CHUNK) n = CHUNK;
    const float* sb = smB + buf * (CHUNK * 64);
    v8f cz = {};
#pragma unroll
    for (int t4 = 0; t4 < CHUNK; t4 += 4) {   // static bound, 4 WMMAs per step
      if (t4 >= n) break;
#pragma unroll
      for (int u = 0; u < 4; ++u) {
        v2f b = ((const v2f*)(sb + (t4 + u) * 64))[lane];
        // 8 args: (neg_a, A, neg_b, B, c_mod, C, reuse_a, reuse_b)
        v8f d = __builtin_amdgcn_wmma_f32_16x16x4_f32(false, a, false, b, (short)0, cz,
                                                      false, false);
#pragma unroll
        for (int k = 0; k < 8; ++k) rmin[k] = fminf(rmin[k], d[k]);
      }
    }
  }
#pragma unroll
  for (int k = 0; k < 8; ++k) {
    float m = rmin[k];
    m = fminf(m, __shfl_xor(m, 1, 32));
    m = fminf(m, __shfl_xor(m, 2, 32));
    m = fminf(m, __shfl_xor(m, 4, 32));
    m = fminf(m, __shfl_xor(m, 8, 32));
    rmin[k] = m;
  }
}

// One wave per A tile, gridDim.y = NSPLIT slices of the (padded) B-tile range.
// numBtiles is a multiple of TILE_ALIGN -> every split gets exactly tps tiles,
// tps % 4 == 0. Writes raw row-min partials to minBuf[row*NSPLIT + split].
// No early return (barrier participation + EXEC all-ones for WMMA):
// overflow waves clamp their tile index and just skip the store.
__global__ void __launch_bounds__(256)
k_min_wmma(const float* __restrict__ Abuf, const float* __restrict__ Bbuf,
           int numAtiles, int numBtiles, float* __restrict__ minBuf) {
  __shared__ float smB[2 * CHUNK * 64];
  int tid = threadIdx.x;
  int wid = tid >> 5, lane = tid & 31;
  int atile0 = blockIdx.x * 8 + wid;
  int atile = (atile0 < numAtiles) ? atile0 : (numAtiles - 1);
  int split = blockIdx.y;
  int tps = numBtiles / NSPLIT;        // exact: numBtiles % (NSPLIT*4) == 0
  int base = split * tps;
  v8f rmin;
  staged_row_min(Abuf, Bbuf, atile, base, tps, smB, lane, tid, rmin);
  if (atile0 < numAtiles && (lane & 15) == 0) {
    int half = lane >> 4;
#pragma unroll
    for (int k = 0; k < 8; ++k) {
      int row = atile0 * 16 + half * 8 + k;
      minBuf[(size_t)row * NSPLIT + split] = rmin[k];
    }
  }
}

// ---------------------------------------------------------------------------
// Combine split-mins. Forward: sum(fp * (min + |a|^2)) into per-wave partials.
// ---------------------------------------------------------------------------
__global__ void k_combine_fwd(const float* __restrict__ minBuf, const float* __restrict__ nA,
                              const float* __restrict__ fp, int Fs,
                              float* __restrict__ fwdPart) {
  int i = blockIdx.x * blockDim.x + threadIdx.x;
  float v = 0.f;
  if (i < Fs) {
    float m = BIG_F;
#pragma unroll
    for (int s = 0; s < NSPLIT; ++s) m = fminf(m, minBuf[(size_t)i * NSPLIT + s]);
    v = fp[i] * (m + nA[i]);
  }
  float s = waveReduceSum(v);
  if ((threadIdx.x & 31) == 0) fwdPart[(blockIdx.x * blockDim.x + threadIdx.x) >> 5] = s;
}

// Reverse: per-sample min dist, unscaled weighted sum partials + global max
// (bitwise u32 atomicMax: order-independent -> deterministic).
__global__ void k_combine_rev(const float* __restrict__ minBuf, const float* __restrict__ nS,
                              const float* __restrict__ fp, int P, int S,
                              float* __restrict__ revPart, unsigned* __restrict__ max_bits) {
  int i = blockIdx.x * blockDim.x + threadIdx.x;
  float v = 0.f, dmax = 0.f;
  if (i < P) {
    float m = BIG_F;
#pragma unroll
    for (int s = 0; s < NSPLIT; ++s) m = fminf(m, minBuf[(size_t)i * NSPLIT + s]);
    float d = m + nS[i];
    v = fp[i / S] * d;
    dmax = fmaxf(d, 0.f);
  }
  float sum = waveReduceSum(v);
  float mx = waveReduceMax(dmax);
  if ((threadIdx.x & 31) == 0) {
    revPart[(blockIdx.x * blockDim.x + threadIdx.x) >> 5] = sum;
    atomicMax(max_bits, __float_as_uint(mx));
  }
}

// Final combine (single thread; deterministic serial sums over ~1.2k partials).
__global__ void k_finalize(const float* __restrict__ fwdPart, int nFwd,
                           const float* __restrict__ revPart, int nRev,
                           const float* __restrict__ fpPart, int nFp,
                           const unsigned* __restrict__ max_bits,
                           int Fs, float* __restrict__ out) {
  float fwd = 0.f;
  for (int i = 0; i < nFwd; ++i) fwd += fwdPart[i];
  float rev = 0.f;
  for (int i = 0; i < nRev; ++i) rev += revPart[i];
  float sfp = 0.f;
  for (int i = 0; i < nFp; ++i) sfp += fpPart[i];
  float maxd = __uint_as_float(max_bits[0]);
  out[0] = fwd + PROB_PEN_F * ((float)Fs - sfp) + rev * (REV_SCALE_F / (maxd + EPS_F));
}

// ---------------------------------------------------------------------------
extern "C" void kernel_launch(void* const* d_in, const int* in_sizes, int n_in,
                              void* d_out, int out_size, void* d_ws, size_t ws_size,
                              hipStream_t stream) {
  (void)n_in; (void)out_size; (void)ws_size;
  const float* origV = (const float*)d_in[0];
  const int*   origF = (const int*)d_in[1];
  const float* simpV = (const float*)d_in[2];
  const int*   simpF = (const int*)d_in[3];
  const float* fp    = (const float*)d_in[4];
  const float* u1    = (const float*)d_in[5];
  const float* u2    = (const float*)d_in[6];
  float* out = (float*)d_out;

  const int No = in_sizes[0] / 3;
  const int Fo = in_sizes[1] / 3;
  const int Fs = in_sizes[3] / 3;
  const int S  = in_sizes[5] / Fs;
  const int P  = Fs * S;

  // B tile counts padded to TILE_ALIGN (sentinel tiles never win the min).
  const int nBtF = ((Fo + 15) / 16 + TILE_ALIGN - 1) / TILE_ALIGN * TILE_ALIGN;
  const int nBtR = ((No + 15) / 16 + TILE_ALIGN - 1) / TILE_ALIGN * TILE_ALIGN;
  const int FoPad = nBtF * 16;
  const int NoPad = nBtR * 16;
  const int nAtF = (Fs + 15) / 16, FsPad = nAtF * 16;   // forward A tiles
  const int nAtR = (P + 15) / 16,  PPad  = nAtR * 16;   // reverse A tiles

  const int blkPackFs = (FsPad + 255) / 256;
  const int blkCmbF   = (FsPad + 255) / 256;
  const int blkCmbR   = (PPad + 255) / 256;
  const int nFpPart   = blkPackFs * 8;
  const int nFwdPart  = blkCmbF * 8;
  const int nRevPart  = blkCmbR * 8;

  float* ws = (float*)d_ws;
  size_t off = 0;
  float* origB = ws + off; off += (size_t)nBtF * 64;
  float* simpA = ws + off; off += (size_t)nAtF * 64;
  float* nA    = ws + off; off += (size_t)FsPad;
  float* vertB = ws + off; off += (size_t)nBtR * 64;
  float* sampA = ws + off; off += (size_t)nAtR * 64;
  float* nS    = ws + off; off += (size_t)PPad;
  float* minF  = ws + off; off += (size_t)FsPad * NSPLIT;
  float* minR  = ws + off; off += (size_t)PPad * NSPLIT;
  float* fwdP  = ws + off; off += (size_t)nFwdPart;
  float* revP  = ws + off; off += (size_t)nRevPart;
  float* fpP   = ws + off; off += (size_t)nFpPart;
  unsigned* maxBits = (unsigned*)(ws + off); off += 16;

  k_init<<<1, 32, 0, stream>>>(maxBits);
  k_pack_orig_bc<<<(FoPad + 255) / 256, 256, 0, stream>>>(origV, origF, Fo, FoPad, origB);
  k_pack_simp_bc<<<blkPackFs, 256, 0, stream>>>(simpV, simpF, fp, Fs, FsPad, simpA, nA, fpP);
  k_pack_verts<<<(NoPad + 255) / 256, 256, 0, stream>>>(origV, No, NoPad, vertB);
  k_pack_samples<<<(PPad + 255) / 256, 256, 0, stream>>>(simpV, simpF, u1, u2, Fs, S, P,
                                                         PPad, sampA, nS);
  dim3 gridF((nAtF + 7) / 8, NSPLIT);
  dim3 gridR((nAtR + 7) / 8, NSPLIT);
  k_min_wmma<<<gridF, 256, 0, stream>>>(simpA, origB, nAtF, nBtF, minF);
  k_min_wmma<<<gridR, 256, 0, stream>>>(sampA, vertB, nAtR, nBtR, minR);
  k_combine_fwd<<<blkCmbF, 256, 0, stream>>>(minF, nA, fp, Fs, fwdP);
  k_combine_rev<<<blkCmbR, 256, 0, stream>>>(minR, nS, fp, P, S, revP, maxBits);
  k_finalize<<<1, 1, 0, stream>>>(fwdP, nFwdPart, revP, nRevPart, fpP, nFpPart,
                                  maxBits, Fs, out);
}